// Attention_16475494547920
// MI455X (gfx1250) — compile-verified
//
#include <hip/hip_runtime.h>
#include <hip/hip_bf16.h>
#include <stdint.h>

typedef __attribute__((ext_vector_type(2))) float v2f;
typedef __attribute__((ext_vector_type(8))) float v8f;

#define HEADS 8
#define DHEAD 64
#define SEQ   1024
#define DIM   512
#define QKV_LD (3 * DIM)     // 1536
#define ATT_SCALE 0.125f     // 64^-0.5
#define LN_EPS 1e-5f
#define JC 64                // attention j-chunk
#define NEG_INF (-3.402823466e38f)

static __device__ __forceinline__ v8f wmma_f32(v2f a, v2f b, v8f c) {
  // V_WMMA_F32_16X16X4_F32 : D(16x16 f32) = A(16x4 f32) * B(4x16 f32) + C
  return __builtin_amdgcn_wmma_f32_16x16x4_f32(false, a, false, b, (short)0, c,
                                               false, false);
}

// CDNA5 async global->LDS copy (ASYNCcnt-tracked, no VGPR round trip).
// lds_off = wave-relative LDS byte address (low 32 bits of a flat LDS pointer).
static __device__ __forceinline__ void async_b128(uint32_t lds_off,
                                                  const float* gptr) {
  asm volatile("global_load_async_to_lds_b128 %0, %1, off"
               :
               : "v"(lds_off), "v"((uint64_t)(uintptr_t)gptr)
               : "memory");
}
static __device__ __forceinline__ void wait_async_le4() {
  asm volatile("s_wait_asynccnt 0x4" ::: "memory");
}
static __device__ __forceinline__ void wait_async_0() {
  asm volatile("s_wait_asynccnt 0x0" ::: "memory");
}

// ---------------------------------------------------------------------------
// fp32 WMMA GEMM with async-LDS double-buffered B staging.
// C[M,N] = A[M,K] @ B[K,N] (+ bias[N]).  N,K compile-time (shift/add addressing).
// Block: 256 threads = 8 waves; block tile 128 rows x 64 cols.
// B chunk [KC=64][64] staged in LDS via global_load_async_to_lds_b128 and
// shared by all 8 waves; next chunk fills while current chunk is consumed.
// A-fragment: lanes 0-15 -> M=lane, K=k0+{0,1}; lanes 16-31 -> M=lane-16, K=k0+{2,3}.
// B-fragment: lanes 0-15 -> N=lane, K=k0+{0,1}; lanes 16-31 -> N=lane-16, K=k0+{2,3}.
// ---------------------------------------------------------------------------
template <int N, int K>
__global__ __launch_bounds__(256) void gemm_f32_wmma_lds(
    const float* __restrict__ A, const float* __restrict__ B,
    const float* __restrict__ bias, float* __restrict__ C, int M) {
  constexpr int KC  = 64;   // k-chunk depth
  constexpr int LDB = 68;   // padded LDS row (floats), 16B-aligned groups kept
  __shared__ float sB[2][KC * LDB];  // 2 x 17 KB

  const int tid    = threadIdx.x;
  const int lane   = tid & 31;
  const int lanelo = lane & 15;
  const int hi     = lane >> 4;
  const int wv     = tid >> 5;

  constexpr int nunits = N >> 6;
  const int brow = blockIdx.x / nunits;
  const int n0   = (blockIdx.x % nunits) << 6;
  const int i0   = brow * 128 + wv * 16;
  if (i0 >= M) return;

  // Each thread issues 4 async b128 transfers per chunk (1024 units total).
  auto issue_chunk = [&](int k0, int buf) {
    const float* src = B + (size_t)k0 * N + n0;
#pragma unroll
    for (int q = 0; q < 4; ++q) {
      const int u  = tid + q * 256;   // b128 unit id
      const int kk = u >> 4;          // row in chunk
      const int c4 = (u & 15) << 2;   // starting column (float)
      const uint32_t lds_off =
          (uint32_t)(uintptr_t)(&sB[buf][kk * LDB + c4]);
      async_b128(lds_off, src + (size_t)kk * N + c4);
    }
  };

  v8f acc[4] = {};
  const float* arow = A + (size_t)(i0 + lanelo) * K + 2 * hi;

  issue_chunk(0, 0);
  constexpr int NCH = K / KC;
  for (int c = 0; c < NCH; ++c) {
    if (c + 1 < NCH) {
      issue_chunk((c + 1) * KC, (c + 1) & 1);
      wait_async_le4();   // chunk c's 4 transfers complete; next 4 in flight
    } else {
      wait_async_0();
    }
    __syncthreads();

    const float* sb = sB[c & 1];
    // Batch-load this chunk's 16 A-fragments (single load clause).
    v2f af[16];
#pragma unroll
    for (int t = 0; t < 16; ++t)
      af[t] = *(const v2f*)(arow + c * KC + 4 * t);

#pragma unroll
    for (int t = 0; t < 16; ++t) {
      const int kk = 4 * t + 2 * hi;
      const float* r0 = sb + kk * LDB + lanelo;
      const float* r1 = r0 + LDB;
#pragma unroll
      for (int nt = 0; nt < 4; ++nt) {
        v2f bf;
        bf.x = r0[nt * 16];
        bf.y = r1[nt * 16];
        acc[nt] = wmma_f32(af[t], bf, acc[nt]);
      }
    }
    __syncthreads();  // all waves done with buf[c&1] before chunk c+2 refills it
  }

#pragma unroll
  for (int nt = 0; nt < 4; ++nt) {
    const int col = n0 + nt * 16 + lanelo;
    const float bv = bias ? bias[col] : 0.0f;
#pragma unroll
    for (int r = 0; r < 8; ++r)
      C[(size_t)(i0 + r + 8 * hi) * N + col] = acc[nt][r] + bv;
  }
}

// ---------------------------------------------------------------------------
// Fused re-attention: softmax(QK^T*s) -> head-mix -> LayerNorm over heads -> @V
// One workgroup per (batch b, 16-query tile). 8 waves = 1 wave per head.
// Pass A: online softmax stats over JC-column chunks (WMMA S-tiles -> LDS).
// Pass B: recompute S, normalize, 8x8 head mix + LN per (i,j), WMMA @V.
// ---------------------------------------------------------------------------
__global__ __launch_bounds__(256) void reattn_fused_kernel(
    const float* __restrict__ qkv, const float* __restrict__ reattn_w,
    const float* __restrict__ ln_g, const float* __restrict__ ln_b,
    float* __restrict__ out) {
  __shared__ float sA[HEADS][16][JC];  // 32 KB scratch (S / reattn)
  __shared__ float sM[HEADS * 16];
  __shared__ float sL[HEADS * 16];
  __shared__ float sW[HEADS * HEADS];
  __shared__ float sG[HEADS], sB2[HEADS];

  const int tid    = threadIdx.x;
  const int wv     = tid >> 5;
  const int lane   = tid & 31;
  const int lanelo = lane & 15;
  const int hi     = lane >> 4;

  const int b  = blockIdx.x >> 6;
  const int i0 = (blockIdx.x & 63) << 4;

  if (tid < 64) sW[tid] = reattn_w[tid];
  if (tid < 8) { sG[tid] = ln_g[tid]; sB2[tid] = ln_b[tid]; }
  if (tid < 128) { sM[tid] = NEG_INF; sL[tid] = 0.0f; }

  const float* qbase = qkv + (size_t)b * SEQ * QKV_LD;
  const float* kbase = qbase + DIM;
  const float* vbase = qbase + 2 * DIM;

  // Preload Q tile (16x64) as 16 A-fragments.
  v2f qf[16];
  {
    const float* qp = qbase + (size_t)(i0 + lanelo) * QKV_LD + wv * DHEAD + 2 * hi;
#pragma unroll
    for (int t = 0; t < 16; ++t) qf[t] = *(const v2f*)(qp + 4 * t);
  }
  __syncthreads();

  // ---------------- Pass A: online softmax statistics ----------------
  for (int jc = 0; jc < SEQ / JC; ++jc) {
    const int jbase = jc * JC;
#pragma unroll
    for (int jt = 0; jt < JC / 16; ++jt) {
      const float* kp = kbase +
          (size_t)(jbase + jt * 16 + lanelo) * QKV_LD + wv * DHEAD + 2 * hi;
      v2f kf[16];
#pragma unroll
      for (int t = 0; t < 16; ++t) kf[t] = *(const v2f*)(kp + 4 * t);
      v8f s = {};
#pragma unroll
      for (int t = 0; t < 16; ++t) s = wmma_f32(qf[t], kf[t], s);
#pragma unroll
      for (int r = 0; r < 8; ++r)
        sA[wv][r + 8 * hi][jt * 16 + lanelo] = s[r] * ATT_SCALE;
    }
    __syncthreads();
    if (tid < 128) {
      const int hh = tid >> 4, ii = tid & 15;
      float m = sM[tid];
      float cmax = NEG_INF;
      for (int j = 0; j < JC; ++j) cmax = fmaxf(cmax, sA[hh][ii][j]);
      const float nm = fmaxf(m, cmax);
      float acc = 0.0f;
      for (int j = 0; j < JC; ++j) acc += __expf(sA[hh][ii][j] - nm);
      sL[tid] = sL[tid] * __expf(m - nm) + acc;
      sM[tid] = nm;
    }
    __syncthreads();
  }
  if (tid < 128) sL[tid] = 1.0f / sL[tid];
  __syncthreads();

  // ---------------- Pass B: recompute, mix, LN, accumulate @V ----------------
  v8f acc[4] = {};
  for (int jc = 0; jc < SEQ / JC; ++jc) {
    const int jbase = jc * JC;
#pragma unroll
    for (int jt = 0; jt < JC / 16; ++jt) {
      const float* kp = kbase +
          (size_t)(jbase + jt * 16 + lanelo) * QKV_LD + wv * DHEAD + 2 * hi;
      v2f kf[16];
#pragma unroll
      for (int t = 0; t < 16; ++t) kf[t] = *(const v2f*)(kp + 4 * t);
      v8f s = {};
#pragma unroll
      for (int t = 0; t < 16; ++t) s = wmma_f32(qf[t], kf[t], s);
#pragma unroll
      for (int r = 0; r < 8; ++r)
        sA[wv][r + 8 * hi][jt * 16 + lanelo] = s[r] * ATT_SCALE;
    }
    __syncthreads();

    float val[4][8];
#pragma unroll
    for (int p = 0; p < 4; ++p) {
      const int pos = tid + p * 256;
      const int ii = pos >> 6, jj = pos & 63;
      float a[8], mix[8];
#pragma unroll
      for (int h = 0; h < 8; ++h)
        a[h] = __expf(sA[h][ii][jj] - sM[h * 16 + ii]) * sL[h * 16 + ii];
      float mean = 0.0f;
#pragma unroll
      for (int g = 0; g < 8; ++g) {
        float s = 0.0f;
#pragma unroll
        for (int h = 0; h < 8; ++h) s += a[h] * sW[h * 8 + g];
        mix[g] = s;
        mean += s;
      }
      mean *= 0.125f;
      float var = 0.0f;
#pragma unroll
      for (int g = 0; g < 8; ++g) { float d = mix[g] - mean; var += d * d; }
      var *= 0.125f;
      const float rn = rsqrtf(var + LN_EPS);
#pragma unroll
      for (int g = 0; g < 8; ++g)
        val[p][g] = (mix[g] - mean) * rn * sG[g] + sB2[g];
    }
    __syncthreads();
#pragma unroll
    for (int p = 0; p < 4; ++p) {
      const int pos = tid + p * 256;
      const int ii = pos >> 6, jj = pos & 63;
#pragma unroll
      for (int g = 0; g < 8; ++g) sA[g][ii][jj] = val[p][g];
    }
    __syncthreads();

    // out_g += reattn_g[16 x JC] @ V_g[JC x 64]
    v2f af[16];
#pragma unroll
    for (int t = 0; t < 16; ++t)
      af[t] = *(const v2f*)&sA[wv][lanelo][4 * t + 2 * hi];
#pragma unroll
    for (int nt = 0; nt < 4; ++nt) {
      v2f vf[16];
#pragma unroll
      for (int t = 0; t < 16; ++t) {
        const float* vp = vbase +
            (size_t)(jbase + 4 * t + 2 * hi) * QKV_LD + wv * DHEAD +
            nt * 16 + lanelo;
        vf[t].x = vp[0];
        vf[t].y = vp[QKV_LD];
      }
#pragma unroll
      for (int t = 0; t < 16; ++t) acc[nt] = wmma_f32(af[t], vf[t], acc[nt]);
    }
    __syncthreads();
  }

  float* obase = out + (size_t)(b * SEQ + i0) * DIM + wv * DHEAD;
#pragma unroll
  for (int nt = 0; nt < 4; ++nt)
#pragma unroll
    for (int r = 0; r < 8; ++r)
      obase[(size_t)(r + 8 * hi) * DIM + nt * 16 + lanelo] = acc[nt][r];
}

// ---------------------------------------------------------------------------
extern "C" void kernel_launch(void* const* d_in, const int* in_sizes, int n_in,
                              void* d_out, int out_size, void* d_ws,
                              size_t ws_size, hipStream_t stream) {
  (void)in_sizes; (void)n_in; (void)out_size; (void)ws_size;
  const float* x        = (const float*)d_in[0];
  const float* w_qkv    = (const float*)d_in[1];
  const float* reattn_w = (const float*)d_in[2];
  const float* ln_g     = (const float*)d_in[3];
  const float* ln_b     = (const float*)d_in[4];
  const float* w_out    = (const float*)d_in[5];
  const float* b_out    = (const float*)d_in[6];
  float* out = (float*)d_out;

  float* qkv   = (float*)d_ws;                 // 8192 x 1536 (48 MB)
  float* attno = qkv + (size_t)8192 * QKV_LD;  // 8192 x 512  (16 MB)

  // qkv = x @ w_qkv : 64 row-blocks x 24 col-units
  gemm_f32_wmma_lds<QKV_LD, DIM><<<64 * 24, 256, 0, stream>>>(
      x, w_qkv, nullptr, qkv, 8192);
  // fused attention: 8 batches x 64 query tiles
  reattn_fused_kernel<<<512, 256, 0, stream>>>(qkv, reattn_w, ln_g, ln_b, attno);
  // out = attno @ w_out + b_out : 64 row-blocks x 8 col-units
  gemm_f32_wmma_lds<DIM, DIM><<<64 * 8, 256, 0, stream>>>(
      attno, w_out, b_out, out, 8192);
}